// RotaryEmbedding_30245159698623
// MI455X (gfx1250) — compile-verified
//
#include <hip/hip_runtime.h>

// RoPE for (B=2, S=4096, D=4096) fp32 q & k with shared freqs_cis table.
// Memory-bound: ~603 MB moved, ~26 us floor at 23.3 TB/s. All traffic is
// 128-bit; x/out streamed with non-temporal hints, freqs staged via the
// gfx1250 async global->LDS path and reused 4x per thread (q/k x b0/b1).

typedef float v4f __attribute__((ext_vector_type(4)));

constexpr int B_ = 2;
constexpr int S_ = 4096;
constexpr int D_ = 4096;
constexpr int D4 = D_ / 4;                       // 1024 float4 per row
constexpr int THREADS = 256;                     // 8 wave32 per block
constexpr long long PER_TENSOR_F4 = (long long)B_ * S_ * D4;  // float4 per tensor

__global__ __launch_bounds__(THREADS) void rope_kernel(
    const float* __restrict__ xq,
    const float* __restrict__ xk,
    const float* __restrict__ fc,
    const int*   __restrict__ start_p,
    float*       __restrict__ out)
{
    __shared__ v4f fc_lds[THREADS];

    const int tid = blockIdx.x * THREADS + threadIdx.x;   // 0 .. S_*D4-1
    const int s   = tid >> 10;                            // position in sequence
    const int d4  = tid & (D4 - 1);                       // float4 index in row
    const int start = start_p[0];                         // uniform scalar load

    // ---- issue async freqs fetch (gfx1250 async-to-LDS, ASYNCcnt) ----------
    const v4f* fcv = reinterpret_cast<const v4f*>(fc);
    const unsigned long long fc_addr =
        (unsigned long long)(const void*)(fcv + (size_t)(start + s) * D4 + d4);
    // Low 32 bits of a generic shared-space address == LDS byte offset.
    unsigned lds_off = (unsigned)(size_t)(&fc_lds[threadIdx.x]);
    asm volatile("global_load_async_to_lds_b128 %0, %1, off"
                 :: "v"(lds_off), "v"(fc_addr)
                 : "memory");

    // ---- stream x while the async copy is in flight (non-temporal) ---------
    const v4f* xqv = reinterpret_cast<const v4f*>(xq);
    const v4f* xkv = reinterpret_cast<const v4f*>(xk);
    const int b1 = S_ * D4;                               // float4 offset of batch 1
    v4f q0 = __builtin_nontemporal_load(xqv + tid);
    v4f q1 = __builtin_nontemporal_load(xqv + tid + b1);
    v4f k0 = __builtin_nontemporal_load(xkv + tid);
    v4f k1 = __builtin_nontemporal_load(xkv + tid + b1);

    // ---- wait for async copy, read own slot (no barrier needed) ------------
    asm volatile("s_wait_asynccnt 0" ::: "memory");
    const v4f f = fc_lds[threadIdx.x];                    // (c0, s0, c1, s1)

    auto rot = [&f](v4f x) {
        v4f r;
        r.x = x.x * f.x - x.y * f.y;   // e0*c0 - o0*s0
        r.y = x.x * f.y + x.y * f.x;   // e0*s0 + o0*c0
        r.z = x.z * f.z - x.w * f.w;   // e1*c1 - o1*s1
        r.w = x.z * f.w + x.w * f.z;   // e1*s1 + o1*c1
        return r;
    };

    // ---- streamed stores (non-temporal, write-once) ------------------------
    v4f* oq = reinterpret_cast<v4f*>(out);
    v4f* ok = oq + PER_TENSOR_F4;                         // k output after q
    __builtin_nontemporal_store(rot(q0), oq + tid);
    __builtin_nontemporal_store(rot(q1), oq + tid + b1);
    __builtin_nontemporal_store(rot(k0), ok + tid);
    __builtin_nontemporal_store(rot(k1), ok + tid + b1);
}

extern "C" void kernel_launch(void* const* d_in, const int* in_sizes, int n_in,
                              void* d_out, int out_size, void* d_ws, size_t ws_size,
                              hipStream_t stream)
{
    (void)in_sizes; (void)n_in; (void)out_size; (void)d_ws; (void)ws_size;
    const float* xq    = (const float*)d_in[0];
    const float* xk    = (const float*)d_in[1];
    const float* fc    = (const float*)d_in[2];
    const int*   start = (const int*)d_in[3];
    float*       out   = (float*)d_out;

    const int total_f4 = S_ * D4;                 // one thread per (s, d4)
    const int blocks   = total_f4 / THREADS;      // 16384
    rope_kernel<<<dim3(blocks), dim3(THREADS), 0, stream>>>(xq, xk, fc, start, out);
}